// MEELoss_35364760715694
// MI455X (gfx1250) — compile-verified
//
#include <hip/hip_runtime.h>
#include <math.h>

typedef __attribute__((ext_vector_type(2))) float v2f;
typedef __attribute__((ext_vector_type(8))) float v8f;

#define THREADS 1024
#define NWAVES (THREADS / 32)

// total = gauss_norm/(N^2) * (sum_i exp(-e_i/(2s^2))) * (sum_j exp(+e_j/(2s^2)))
__global__ __launch_bounds__(THREADS)
void mee_loss_kernel(const float* __restrict__ y_pred,
                     const float* __restrict__ y_true,
                     const float* __restrict__ kw,
                     float* __restrict__ out,
                     int n) {
    const float sigma   = kw[0];
    const float inv2s2  = 1.0f / (2.0f * sigma * sigma);
    const float gnorm   = 1.0f / (sqrtf(2.0f * 3.14159f) * sigma);
    const float normcst = 1.0f / ((float)n * (float)n);

    // Per-thread strided accumulation of the two exp sums.
    float pneg = 0.0f, ppos = 0.0f;
    for (int i = threadIdx.x; i < n; i += THREADS) {
        float e = fabsf(y_true[i] - y_pred[i]);
        float a = e * inv2s2;
        pneg += __expf(-a);
        ppos += __expf(a);
    }

    // wave32 butterfly reduce
    #pragma unroll
    for (int m = 16; m >= 1; m >>= 1) {
        pneg += __shfl_xor(pneg, m, 32);
        ppos += __shfl_xor(ppos, m, 32);
    }

    __shared__ float s_neg[NWAVES];
    __shared__ float s_pos[NWAVES];
    const int lane = threadIdx.x & 31;
    const int wid  = threadIdx.x >> 5;
    if (lane == 0) { s_neg[wid] = pneg; s_pos[wid] = ppos; }
    __syncthreads();

    if (wid == 0) {  // wave-uniform branch: EXEC all-ones inside
        // Pack 32 neg partials into rows 0..7 (4 K-slots each) and
        // 32 pos partials into rows 8..15 of the 16x4 f32 A matrix.
        // f32 A layout: lane L (L<16):  row=L,    a.x->K=0, a.y->K=1
        //               lane L (L>=16): row=L-16, a.x->K=2, a.y->K=3
        const int half = lane >> 4;   // which K pair this lane feeds
        const int m    = lane & 15;   // row index
        v2f a;
        if (m < 8) {
            int base = 4 * m + 2 * half;           // neg[base], neg[base+1]
            a.x = s_neg[base]; a.y = s_neg[base + 1];
        } else {
            int base = 4 * (m - 8) + 2 * half;     // pos[base], pos[base+1]
            a.x = s_pos[base]; a.y = s_pos[base + 1];
        }
        v2f b; b.x = 1.0f; b.y = 1.0f;             // B = ones(4x16)
        v8f c = {};
        // D[m][n] = sum_k A[m][k] : row-sums of the packed partials.
        c = __builtin_amdgcn_wmma_f32_16x16x4_f32(
                /*neg_a=*/false, a, /*neg_b=*/false, b,
                /*c_mod=*/(short)0, c, /*reuse_a=*/false, /*reuse_b=*/false);

        // D VGPR j: lanes 0-15 hold row j (neg rows 0..7),
        //           lanes 16-31 hold row j+8 (pos rows 8..15).
        float s = c[0] + c[1] + c[2] + c[3] + c[4] + c[5] + c[6] + c[7];
        // lanes 0-15: total_neg ; lanes 16-31: total_pos
        float other = __shfl_xor(s, 16, 32);
        if (lane == 0) {
            out[0] = gnorm * normcst * s * other;
        }
    }
}

extern "C" void kernel_launch(void* const* d_in, const int* in_sizes, int n_in,
                              void* d_out, int out_size, void* d_ws, size_t ws_size,
                              hipStream_t stream) {
    const float* y_pred = (const float*)d_in[0];
    const float* y_true = (const float*)d_in[1];
    const float* kw     = (const float*)d_in[2];
    float* out = (float*)d_out;
    const int n = in_sizes[0];

    mee_loss_kernel<<<1, THREADS, 0, stream>>>(y_pred, y_true, kw, out, n);
}